// BinaryMemoryRNNCell_56873956934278
// MI455X (gfx1250) — compile-verified
//
#include <hip/hip_runtime.h>
#include <hip/hip_bf16.h>

// ---- problem constants (match reference) ----
#define HID   1024
#define BATCH 128
#define NBITS 10

typedef __attribute__((ext_vector_type(2))) float v2f;
typedef __attribute__((ext_vector_type(8))) float v8f;

// -------------------------------------------------------------------------
// Kernel 1: logits = h_prev @ M_w^T + M_b ; pack hard-threshold bits into
// two 10-bit indices per batch row.  One block per batch row, 20 waves,
// each wave computes one logit via a wave32 shuffle reduction.
// -------------------------------------------------------------------------
__global__ __launch_bounds__(640) void idx_kernel(const float* __restrict__ h_prev,
                                                  const float* __restrict__ Mw,
                                                  const float* __restrict__ Mb,
                                                  int* __restrict__ idx) {
    __shared__ float sl[2 * NBITS];
    const int b    = blockIdx.x;
    const int wave = threadIdx.x >> 5;   // 0..19 -> logit index
    const int lane = threadIdx.x & 31;

    const float* hrow = h_prev + (size_t)b * HID;
    const float* mrow = Mw + (size_t)wave * HID;
    float acc = 0.0f;
    for (int k = lane; k < HID; k += 32)
        acc += hrow[k] * mrow[k];
#pragma unroll
    for (int off = 16; off > 0; off >>= 1)
        acc += __shfl_xor(acc, off, 32);
    if (lane == 0) sl[wave] = acc + Mb[wave];
    __syncthreads();

    if (threadIdx.x == 0) {
        int i1 = 0, i2 = 0;
#pragma unroll
        for (int j = 0; j < NBITS; ++j) {
            i1 = (i1 << 1) | (sl[j]         > 0.0f ? 1 : 0);  // sigmoid>0.5 <=> logit>0
            i2 = (i2 << 1) | (sl[NBITS + j] > 0.0f ? 1 : 0);
        }
        idx[b]         = i1;   // already in [0, 1023]
        idx[BATCH + b] = i2;
    }
}

// -------------------------------------------------------------------------
// Kernel 2: per-batch gather from mem_tensor into compact [B,H] buffers.
// grid (BATCH, 2), 256 threads, float4 coalesced copy of one 4KB row.
// -------------------------------------------------------------------------
__global__ __launch_bounds__(256) void gather_kernel(const float* __restrict__ mem,
                                                     const int* __restrict__ idx,
                                                     float* __restrict__ hr,
                                                     float* __restrict__ hl) {
    const int b     = blockIdx.x;
    const int which = blockIdx.y;
    const int t     = idx[which * BATCH + b];
    const float4* src = (const float4*)(mem + ((size_t)t * BATCH + b) * HID);
    float4*       dst = (float4*)((which ? hl : hr) + (size_t)b * HID);
    dst[threadIdx.x] = src[threadIdx.x];   // 256 * 16B = 4KB row
}

// -------------------------------------------------------------------------
// Kernel 3: fused 4-way GEMM, full fp32 via V_WMMA_F32_16X16X4_F32.
//   pre = x@Ww^T + h@Uw^T + hr@Qrw^T + hl@Qlw^T      (biases added later)
// Grid: 64 blocks = one per 16 output columns (weight-stationary strip:
// every weight byte is read exactly once chip-wide).  8 waves per block,
// wave w owns batch tile [16w, 16w+16).
//
// Four INDEPENDENT accumulator chains (one per GEMM) so the matrix pipe
// is not serialized by C-operand RAW hazards; summed elementwise at the end.
//
// f32 WMMA fragment layout (ISA 7.12.2):
//   A 16x4 : lane m (0-15) holds row m; VGPR0/1 = K 2*half+0 / 2*half+1
//   B 4x16 : lane n (0-15) holds col n; VGPR0/1 = K 2*half+0 / 2*half+1
//   C 16x16: VGPR r = row r + 8*half, col = lane&15
// -------------------------------------------------------------------------
__global__ __launch_bounds__(256) void gemm_kernel(const float* __restrict__ x,
                                                   const float* __restrict__ h,
                                                   const float* __restrict__ hr,
                                                   const float* __restrict__ hl,
                                                   const float* __restrict__ Ww,
                                                   const float* __restrict__ Uw,
                                                   const float* __restrict__ Qrw,
                                                   const float* __restrict__ Qlw,
                                                   float* __restrict__ pre) {
    const int lane = threadIdx.x & 31;
    const int wave = threadIdx.x >> 5;   // 0..7 -> batch tile
    const int half = lane >> 4;          // K sub-pair select
    const int lid  = lane & 15;

    const int o0   = blockIdx.x * 16;    // output-column strip
    const int arow = wave * 16 + lid;    // batch row fed by this lane (A)
    const int brow = o0 + lid;           // weight row fed by this lane (B = W^T tile)

    const float* ax_p = x   + (size_t)arow * HID + 2 * half;
    const float* ah_p = h   + (size_t)arow * HID + 2 * half;
    const float* ar_p = hr  + (size_t)arow * HID + 2 * half;
    const float* al_p = hl  + (size_t)arow * HID + 2 * half;
    const float* bw_p = Ww  + (size_t)brow * HID + 2 * half;
    const float* bu_p = Uw  + (size_t)brow * HID + 2 * half;
    const float* br_p = Qrw + (size_t)brow * HID + 2 * half;
    const float* bl_p = Qlw + (size_t)brow * HID + 2 * half;

    v8f accW = {};
    v8f accU = {};
    v8f accR = {};
    v8f accL = {};
#pragma unroll 4
    for (int k0 = 0; k0 < HID; k0 += 4) {
        v2f ax = *(const v2f*)(ax_p + k0);
        v2f ah = *(const v2f*)(ah_p + k0);
        v2f ar = *(const v2f*)(ar_p + k0);
        v2f al = *(const v2f*)(al_p + k0);
        v2f bw = *(const v2f*)(bw_p + k0);
        v2f bu = *(const v2f*)(bu_p + k0);
        v2f br = *(const v2f*)(br_p + k0);
        v2f bl = *(const v2f*)(bl_p + k0);
        accW = __builtin_amdgcn_wmma_f32_16x16x4_f32(false, ax, false, bw, (short)0, accW, false, false);
        accU = __builtin_amdgcn_wmma_f32_16x16x4_f32(false, ah, false, bu, (short)0, accU, false, false);
        accR = __builtin_amdgcn_wmma_f32_16x16x4_f32(false, ar, false, br, (short)0, accR, false, false);
        accL = __builtin_amdgcn_wmma_f32_16x16x4_f32(false, al, false, bl, (short)0, accL, false, false);
    }

    const v8f acc = (accW + accU) + (accR + accL);

    const int col = o0 + lid;
#pragma unroll
    for (int r = 0; r < 8; ++r) {
        const int row = wave * 16 + 8 * half + r;     // C layout: VGPR r -> M = r + 8*half
        pre[(size_t)row * HID + col] = acc[r];
    }
}

// -------------------------------------------------------------------------
// Kernel 4: add the four biases, LayerNorm over H per batch row, sigmoid.
// One block per batch row, 256 threads, 4 elements each, LDS tree reduce.
// -------------------------------------------------------------------------
__global__ __launch_bounds__(256) void ln_kernel(const float* __restrict__ pre,
                                                 const float* __restrict__ Wb,
                                                 const float* __restrict__ Ub,
                                                 const float* __restrict__ Qrb,
                                                 const float* __restrict__ Qlb,
                                                 const float* __restrict__ g,
                                                 const float* __restrict__ bb,
                                                 float* __restrict__ out) {
    __shared__ float s1[256];
    __shared__ float s2[256];
    const int b = blockIdx.x;
    const int t = threadIdx.x;

    float vals[4];
    float sum = 0.0f, sq = 0.0f;
#pragma unroll
    for (int i = 0; i < 4; ++i) {
        const int o = t + i * 256;
        const float v = pre[(size_t)b * HID + o] + Wb[o] + Ub[o] + Qrb[o] + Qlb[o];
        vals[i] = v;
        sum += v;
        sq  += v * v;
    }
    s1[t] = sum;
    s2[t] = sq;
    __syncthreads();
    for (int off = 128; off > 0; off >>= 1) {
        if (t < off) { s1[t] += s1[t + off]; s2[t] += s2[t + off]; }
        __syncthreads();
    }
    const float mu  = s1[0] * (1.0f / HID);
    const float var = s2[0] * (1.0f / HID) - mu * mu;
    const float rs  = rsqrtf(var + 1e-5f);
#pragma unroll
    for (int i = 0; i < 4; ++i) {
        const int o = t + i * 256;
        const float nv = (vals[i] - mu) * rs * g[o] + bb[o];
        out[(size_t)b * HID + o] = 1.0f / (1.0f + __expf(-nv));
    }
}

// -------------------------------------------------------------------------
extern "C" void kernel_launch(void* const* d_in, const int* in_sizes, int n_in,
                              void* d_out, int out_size, void* d_ws, size_t ws_size,
                              hipStream_t stream) {
    const float* x   = (const float*)d_in[0];
    const float* hp  = (const float*)d_in[1];
    const float* mem = (const float*)d_in[2];
    const float* Ww  = (const float*)d_in[3];
    const float* Wb  = (const float*)d_in[4];
    const float* Uw  = (const float*)d_in[5];
    const float* Ub  = (const float*)d_in[6];
    const float* Mw  = (const float*)d_in[7];
    const float* Mb  = (const float*)d_in[8];
    const float* Qrw = (const float*)d_in[9];
    const float* Qrb = (const float*)d_in[10];
    const float* Qlw = (const float*)d_in[11];
    const float* Qlb = (const float*)d_in[12];
    const float* lng = (const float*)d_in[13];
    const float* lnb = (const float*)d_in[14];
    float* out = (float*)d_out;

    // workspace layout: [ idx: 2*128 ints (pad to 256 floats) | hr | hl | pre ]
    int*   idx = (int*)d_ws;
    float* hr  = (float*)d_ws + 256;
    float* hl  = hr + (size_t)BATCH * HID;
    float* pre = hl + (size_t)BATCH * HID;

    idx_kernel   <<<BATCH, 640, 0, stream>>>(hp, Mw, Mb, idx);
    gather_kernel<<<dim3(BATCH, 2), 256, 0, stream>>>(mem, idx, hr, hl);
    gemm_kernel  <<<HID / 16, 256, 0, stream>>>(x, hp, hr, hl, Ww, Uw, Qrw, Qlw, pre);
    ln_kernel    <<<BATCH, 256, 0, stream>>>(pre, Wb, Ub, Qrb, Qlb, lng, lnb, out);
}